// SimpleHeadAttention_5634997093266
// MI455X (gfx1250) — compile-verified
//
#include <hip/hip_runtime.h>

typedef __attribute__((ext_vector_type(2))) float v2f;
typedef __attribute__((ext_vector_type(8))) float v8f;

#define B_  4
#define N_  1024
#define C_  1024
#define H_  16
#define HD_ 64

__device__ __forceinline__ v8f wmma_f32(v2f a, v2f b, v8f c) {
  // D = A(16x4, f32) * B(4x16, f32) + C(16x16, f32); fp32-exact path on CDNA5
  return __builtin_amdgcn_wmma_f32_16x16x4_f32(
      /*neg_a=*/false, a, /*neg_b=*/false, b,
      /*c_mod=*/(short)0, c, /*reuse_a=*/false, /*reuse_b=*/false);
}

// Generic row-major GEMM: C[m,n] = sum_k A[m,k]*B[k,n] (+ bias[n])
// Wave computes a 32x64 strip (2 M-tiles x 4 N-tiles = 8 accumulators) so each
// B fragment is reused by two WMMAs. Block = 4 waves = 128 rows x 64 cols.
// Batched over grid.z: z -> (bb = z/Hdiv, hh = z%Hdiv) with independent strides,
// so the same kernel serves plain GEMMs (Hdiv=1) and per-(b,h) attn@V.
__global__ __launch_bounds__(128) void gemm_wmma_f32(
    const float* __restrict__ A, long lda, long sAz,
    const float* __restrict__ Bm, long ldb, long sBb, long sBh,
    const float* __restrict__ bias,
    float* __restrict__ Cm, long ldc, long sCb, long sCh,
    int K, int Hdiv)
{
  const int z  = blockIdx.z;
  const int bb = z / Hdiv;
  const int hh = z % Hdiv;
  A  += (long)z  * sAz;
  Bm += (long)bb * sBb + (long)hh * sBh;
  Cm += (long)bb * sCb + (long)hh * sCh;

  const int wave = threadIdx.x >> 5;
  const int lane = threadIdx.x & 31;
  const int l15  = lane & 15;
  const int lh   = lane >> 4;

  const long m0 = ((long)blockIdx.y * 4 + wave) * 32;
  const long n0 = (long)blockIdx.x * 64;

  v8f acc[2][4] = {};
  const float* arow0 = A + (m0 + l15) * lda;
  const float* arow1 = arow0 + 16 * lda;

  for (int k0 = 0; k0 < K; k0 += 4) {
    const int kk = k0 + 2 * lh;              // lanes 0-15: K={k0,k0+1}; 16-31: K={k0+2,k0+3}
    v2f a0 = *(const v2f*)(arow0 + kk);      // contiguous, 8B-aligned (lda,kk even)
    v2f a1 = *(const v2f*)(arow1 + kk);
#pragma unroll
    for (int t = 0; t < 4; ++t) {
      const long n = n0 + t * 16 + l15;
      v2f b;
      b.x = Bm[(long)kk       * ldb + n];
      b.y = Bm[(long)(kk + 1) * ldb + n];
      acc[0][t] = wmma_f32(a0, b, acc[0][t]);
      acc[1][t] = wmma_f32(a1, b, acc[1][t]);
    }
  }

#pragma unroll
  for (int t = 0; t < 4; ++t) {
    const long n  = n0 + t * 16 + l15;
    const float bv = bias ? bias[n] : 0.0f;
#pragma unroll
    for (int mt = 0; mt < 2; ++mt) {
#pragma unroll
      for (int r = 0; r < 8; ++r) {
        const long m = m0 + mt * 16 + r + 8 * lh;   // C/D layout: lane=col, rows in VGPRs
        Cm[m * ldc + n] = acc[mt][t][r] + bv;
      }
    }
  }
}

// Scores + bias + mask + softmax, writing normalized attn to global.
// One block per (b, h, 16-row slab). 8 waves x (8 tiles of 16 cols) = full 1024 columns,
// so the row softmax reduces entirely within the block.
__global__ __launch_bounds__(256) void attn_softmax_wmma(
    const float* __restrict__ qkv, const float* __restrict__ im,
    const float* __restrict__ mask, float* __restrict__ attn)
{
  __shared__ float red[2][8][16];            // [max|sum][wave][row]

  const int m0 = blockIdx.x * 16;
  const int h  = blockIdx.y;
  const int bb = blockIdx.z;
  const int wave = threadIdx.x >> 5;
  const int lane = threadIdx.x & 31;
  const int l15  = lane & 15;
  const int lh   = lane >> 4;

  // q row for this lane's A-fragment; k base for B-fragments (S = q . k^T => B[kd][n] = K[n][kd])
  const float* qrow  = qkv + (long)(bb * N_ + m0 + l15) * (3 * C_) + h * HD_;
  const float* kbase = qkv + (long)bb * N_ * (3 * C_) + C_ + h * HD_;

  v8f acc[8] = {};
  for (int k0 = 0; k0 < HD_; k0 += 4) {
    const int kk = k0 + 2 * lh;
    v2f a = *(const v2f*)(qrow + kk);        // contiguous along head_dim
#pragma unroll
    for (int t = 0; t < 8; ++t) {
      const int n = wave * 128 + t * 16 + l15;
      v2f b = *(const v2f*)(kbase + (long)n * (3 * C_) + kk);
      acc[t] = wmma_f32(a, b, acc[t]);
    }
  }

  const float scale = 0.125f;                // HD^-0.5, folded post-GEMM (q*scale . k)
  const float* imb = im   + ((long)bb * H_ + h) * N_ * (long)N_;
  const float* mkb = mask + (long)bb * N_ * (long)N_;   // mask is [B,1,N,N]

  float rmax[8];
#pragma unroll
  for (int r = 0; r < 8; ++r) rmax[r] = -3.0e38f;

#pragma unroll
  for (int t = 0; t < 8; ++t) {
    const long n = wave * 128 + t * 16 + l15;
#pragma unroll
    for (int r = 0; r < 8; ++r) {
      const long m = m0 + r + 8 * lh;
      float s = acc[t][r] * scale + imb[m * N_ + n]
              + (1.0f - mkb[m * N_ + n]) * -1.0e9f;
      acc[t][r] = s;
      rmax[r] = fmaxf(rmax[r], s);
    }
  }

  // reduce across the 16 lanes holding the same row (xor stays inside each half-wave)
#pragma unroll
  for (int r = 0; r < 8; ++r)
#pragma unroll
    for (int off = 8; off >= 1; off >>= 1)
      rmax[r] = fmaxf(rmax[r], __shfl_xor(rmax[r], off, 32));

  if (l15 == 0) {
#pragma unroll
    for (int r = 0; r < 8; ++r) red[0][wave][r + 8 * lh] = rmax[r];
  }
  __syncthreads();
#pragma unroll
  for (int r = 0; r < 8; ++r) {
    float v = red[0][0][r + 8 * lh];
#pragma unroll
    for (int w = 1; w < 8; ++w) v = fmaxf(v, red[0][w][r + 8 * lh]);
    rmax[r] = v;
  }

  float rsum[8];
#pragma unroll
  for (int r = 0; r < 8; ++r) rsum[r] = 0.0f;
#pragma unroll
  for (int t = 0; t < 8; ++t)
#pragma unroll
    for (int r = 0; r < 8; ++r) {
      float e = __expf(acc[t][r] - rmax[r]);
      acc[t][r] = e;
      rsum[r] += e;
    }
#pragma unroll
  for (int r = 0; r < 8; ++r)
#pragma unroll
    for (int off = 8; off >= 1; off >>= 1)
      rsum[r] += __shfl_xor(rsum[r], off, 32);

  if (l15 == 0) {
#pragma unroll
    for (int r = 0; r < 8; ++r) red[1][wave][r + 8 * lh] = rsum[r];
  }
  __syncthreads();
#pragma unroll
  for (int r = 0; r < 8; ++r) {
    float v = 0.0f;
#pragma unroll
    for (int w = 0; w < 8; ++w) v += red[1][w][r + 8 * lh];
    rsum[r] = 1.0f / v;
  }

  float* ab = attn + ((long)bb * H_ + h) * N_ * (long)N_;
#pragma unroll
  for (int t = 0; t < 8; ++t) {
    const long n = wave * 128 + t * 16 + l15;
#pragma unroll
    for (int r = 0; r < 8; ++r) {
      const long m = m0 + r + 8 * lh;
      ab[m * N_ + n] = acc[t][r] * rsum[r];
    }
  }
}

extern "C" void kernel_launch(void* const* d_in, const int* in_sizes, int n_in,
                              void* d_out, int out_size, void* d_ws, size_t ws_size,
                              hipStream_t stream) {
  (void)in_sizes; (void)n_in; (void)out_size; (void)ws_size;

  const float* x     = (const float*)d_in[0];   // [B,N,C]
  const float* im    = (const float*)d_in[1];   // [B,H,N,N]
  const float* mask  = (const float*)d_in[2];   // [B,1,N,N]
  const float* Wqkv  = (const float*)d_in[3];   // [C,3C]
  const float* bqkv  = (const float*)d_in[4];   // [3C]
  const float* Wproj = (const float*)d_in[5];   // [C,C]
  const float* bproj = (const float*)d_in[6];   // [C]

  float* out  = (float*)d_out;                         // [B,N,C]
  float* attn = out + (long)B_ * N_ * C_;              // [B,H,N,N]
  float* qkv  = (float*)d_ws;                          // [B,N,3C]  48 MB
  float* ctx  = qkv + (long)B_ * N_ * 3 * C_;          // [B,N,C]   16 MB

  // 1) qkv = x @ W_qkv + b_qkv        (M=4096, N=3072, K=1024)
  gemm_wmma_f32<<<dim3(3 * C_ / 64, (B_ * N_) / 128, 1), 128, 0, stream>>>(
      x, C_, 0, Wqkv, 3 * C_, 0, 0, bqkv, qkv, 3 * C_, 0, 0, C_, 1);

  // 2) attn = softmax(q k^T * scale + int_matrix + maskterm)  -> d_out attn region
  attn_softmax_wmma<<<dim3(N_ / 16, H_, B_), 256, 0, stream>>>(qkv, im, mask, attn);

  // 3) ctx[b,:,h,:] = attn[b,h] @ v[b,h]   (per (b,h): M=1024, N=64, K=1024)
  gemm_wmma_f32<<<dim3(HD_ / 64, N_ / 128, B_ * H_), 128, 0, stream>>>(
      attn, N_, (long)N_ * N_,
      qkv + 2 * C_, 3 * C_, (long)N_ * 3 * C_, HD_,
      nullptr,
      ctx, C_, (long)N_ * C_, HD_,
      N_, H_);

  // 4) out = ctx @ W_proj + b_proj    (M=4096, N=1024, K=1024)
  gemm_wmma_f32<<<dim3(C_ / 64, (B_ * N_) / 128, 1), 128, 0, stream>>>(
      ctx, C_, 0, Wproj, C_, 0, 0, bproj, out, C_, 0, 0, C_, 1);
}